// OAT_44109314130320
// MI455X (gfx1250) — compile-verified
//
#include <hip/hip_runtime.h>
#include <math.h>

// Problem constants (match reference)
#define B_   8
#define C_   128
#define T_   500
#define D_   128
#define H_   8
#define DH_  16      // D_/H_
#define O_   21
#define KW_  9
#define KDIM_ 189    // O_*KW_
#define KPAD_ 192    // KDIM_ padded to multiple of 4
#define TT_  100     // t-tile for fused layer kernel (5 tiles cover 500)

typedef float v2f __attribute__((ext_vector_type(2)));
typedef float v8f __attribute__((ext_vector_type(8)));

__device__ __forceinline__ float selu_f(float x) {
  const float lam = 1.0507009873554805f;
  const float al  = 1.6732632423543772f;
  return x > 0.0f ? lam * x : lam * al * (expf(x) - 1.0f);
}

// ---------------------------------------------------------------------------
// Q = selu(conv1d_same(x[:,occ], wq_l))  as im2col GEMM via V_WMMA_F32_16X16X4_F32
// A = wq_l (128 x 189), B = im2col(x_occ) (189 x 500), C tile = 16x16 f32.
// One wave per (b, mtile, ntile); EXEC all ones around the WMMAs.
// K loop fully unrolled: j0 is compile-time, so the o=j/9, k=j%9 im2col
// indexing folds into immediate ds_load offsets.
// ---------------------------------------------------------------------------
__global__ __launch_bounds__(32) void qconv_wmma(
    const float* __restrict__ xin, const int* __restrict__ occ,
    const float* __restrict__ wq_l, float* __restrict__ Qout)
{
  __shared__ __align__(16) float As[16][KPAD_]; // A tile rows (zero-padded K)
  __shared__ __align__(16) float Bs[22][24];    // x_occ window: 21 rows + zero row
  const int lane  = threadIdx.x;
  const int ntile = blockIdx.x;     // 32 tiles of 16 cover T=500 (padded)
  const int mtile = blockIdx.y;     // 8 tiles of 16 cover D=128
  const int b     = blockIdx.z;
  const int nbase = ntile * 16;

  for (int idx = lane; idx < 16 * KPAD_; idx += 32) {
    int r = idx / KPAD_, j = idx % KPAD_;
    As[r][j] = (j < KDIM_) ? wq_l[(mtile * 16 + r) * KDIM_ + j] : 0.0f;
  }
  for (int idx = lane; idx < 22 * 24; idx += 32) {
    int o = idx / 24, kc = idx % 24;
    int ts = nbase + kc - 4;                    // SAME padding: t + k - 4
    float v = 0.0f;
    if (o < O_ && ts >= 0 && ts < T_) v = xin[(b * C_ + occ[o]) * T_ + ts];
    Bs[o][kc] = v;
  }
  __syncthreads();

  const int lo = lane & 15;
  const int hi = lane >> 4;
  v8f acc = {};
  #pragma unroll
  for (int kk = 0; kk < KPAD_ / 4; ++kk) {
    // f32 A 16x4 layout: lane -> M, half-wave selects K pair, 2 VGPRs = K in pair
    const int j0 = kk * 4 + 2 * hi;
    v2f a, bv;
    a.x = As[lo][j0];
    a.y = As[lo][j0 + 1];
    // f32 B 4x16 layout: lane -> N, same K split
    const int o0 = j0 / 9,        k0 = j0 - o0 * 9;
    const int o1 = (j0 + 1) / 9,  k1 = (j0 + 1) - o1 * 9;
    bv.x = Bs[o0][lo + k0];
    bv.y = Bs[o1][lo + k1];
    acc = __builtin_amdgcn_wmma_f32_16x16x4_f32(
        /*neg_a=*/false, a, /*neg_b=*/false, bv,
        /*c_mod=*/(short)0, acc, /*reuse_a=*/false, /*reuse_b=*/false);
  }

  // C/D 16x16 f32 layout: VGPR r -> row r (lanes 0-15) / row r+8 (lanes 16-31)
  int col = nbase + lo;
  if (col < T_) {
    #pragma unroll
    for (int r = 0; r < 8; ++r) {
      int row = mtile * 16 + r + 8 * hi;
      Qout[(b * D_ + row) * T_ + col] = selu_f(acc[r]);
    }
  }
}

// ---------------------------------------------------------------------------
// Fused per-layer kernel: recomputes the depthwise K/V convs on the fly
// (never materializing the 262MB K/V tensors), computes gates, gated
// projection, and residual update.  One block per (ttile, c, b).
// The 51.2KB Q tile is staged memory->LDS with CDNA5 async-LDS loads
// (global_load_async_to_lds_b128, tracked by ASYNCcnt), bypassing VGPRs.
// ---------------------------------------------------------------------------
__global__ __launch_bounds__(128) void layer_fused(
    const float* __restrict__ xin, const float* __restrict__ Qbuf,
    const float* __restrict__ wk_l, const float* __restrict__ wv_l,
    const float* __restrict__ wp_l, float* __restrict__ xout)
{
  __shared__ __align__(16) float Qs[D_ * TT_];   // 51.2 KB
  __shared__ __align__(16) float wks[D_ * KW_];  // 4.6 KB
  __shared__ __align__(16) float wvs[D_ * KW_];  // 4.6 KB
  __shared__ __align__(16) float wps[D_];
  const int tid   = threadIdx.x;
  const int ttile = blockIdx.x;
  const int c     = blockIdx.y;
  const int b     = blockIdx.z;
  const int tbase = ttile * TT_;

  // Async-stage Q tile: 12800 floats = 3200 x b128, 25 chunks per lane.
  // Global chunks never cross a Q row (TT_=100 and offsets are multiples of 4),
  // and both sides are 16B aligned.
  {
    const float* qbase = Qbuf + (size_t)b * D_ * T_ + tbase;
    #pragma unroll 1
    for (int i = 0; i < 25; ++i) {
      int q   = tid + i * 128;        // chunk index < 3200
      int idx = q * 4;                // float index into Qs
      int j   = idx / TT_;
      int tt  = idx - j * TT_;
      unsigned int lds_off = (unsigned int)(size_t)(const void*)&Qs[idx];
      const float* gp = qbase + j * T_ + tt;
      asm volatile("global_load_async_to_lds_b128 %0, %1, off"
                   :: "v"(lds_off), "v"(gp) : "memory");
    }
  }
  for (int idx = tid; idx < D_ * KW_; idx += 128) {
    wks[idx] = wk_l[c * D_ * KW_ + idx];
    wvs[idx] = wv_l[c * D_ * KW_ + idx];
  }
  if (tid < D_) wps[tid] = wp_l[c * D_ + tid];
  asm volatile("s_wait_asynccnt 0x0" ::: "memory");
  __syncthreads();

  if (tid < TT_) {
    const int t = tbase + tid;                  // always < 500 (5*100 tiles)
    const float* xrow = xin + (b * C_ + c) * T_;
    float xwin[KW_];
    #pragma unroll
    for (int k = 0; k < KW_; ++k) {
      int ts = t + k - 4;
      xwin[k] = (ts >= 0 && ts < T_) ? xrow[ts] : 0.0f;
    }
    float out = 0.0f;
    #pragma unroll
    for (int h = 0; h < H_; ++h) {
      float qkh = 0.0f, vwh = 0.0f;
      #pragma unroll 4
      for (int dd = 0; dd < DH_; ++dd) {
        int j = h * DH_ + dd;
        float ka = 0.0f, va = 0.0f;
        #pragma unroll
        for (int k = 0; k < KW_; ++k) {
          ka += xwin[k] * wks[j * KW_ + k];
          va += xwin[k] * wvs[j * KW_ + k];
        }
        float kj = selu_f(ka);
        float vj = selu_f(va);
        qkh += Qs[j * TT_ + tid] * kj;
        vwh += vj * wps[j];
      }
      float g = selu_f(qkh * 0.25f);            // scale = (D/H)^-0.5 = 0.25
      out += g * vwh;
    }
    out = selu_f(out);
    xout[(b * C_ + c) * T_ + t] = (xwin[4] + out) * 0.70710678118654752f;
  }
}

// ---------------------------------------------------------------------------
// logits[b,u] = sum_{idx<64000} x[b,idx] * w_head[u,idx] + b_head[u]
// Deterministic block-per-u reduction (no float atomics).  w_head rows are
// streamed once; prefetch ahead (lowers to global_prefetch_b8).
// ---------------------------------------------------------------------------
__global__ __launch_bounds__(256) void head_gemm(
    const float* __restrict__ xfin, const float* __restrict__ wh,
    const float* __restrict__ bh, float* __restrict__ out)
{
  __shared__ float red[256];
  const int u   = blockIdx.x;
  const int tid = threadIdx.x;
  float acc[B_];
  #pragma unroll
  for (int b = 0; b < B_; ++b) acc[b] = 0.0f;
  const float* wrow = wh + (size_t)u * (C_ * T_);
  for (int idx = tid; idx < C_ * T_; idx += 256) {
    if (idx + 2048 < C_ * T_) __builtin_prefetch(&wrow[idx + 2048], 0, 1);
    float w = wrow[idx];
    #pragma unroll
    for (int b = 0; b < B_; ++b) acc[b] += w * xfin[b * (C_ * T_) + idx];
  }
  for (int b = 0; b < B_; ++b) {
    red[tid] = acc[b];
    __syncthreads();
    for (int s = 128; s > 0; s >>= 1) {
      if (tid < s) red[tid] += red[tid + s];
      __syncthreads();
    }
    if (tid == 0) out[b * 40 + u] = red[0] + bh[u];
    __syncthreads();
  }
}

// ---------------------------------------------------------------------------
extern "C" void kernel_launch(void* const* d_in, const int* in_sizes, int n_in,
                              void* d_out, int out_size, void* d_ws, size_t ws_size,
                              hipStream_t stream) {
  const float* x     = (const float*)d_in[0];
  const int*   occ   = (const int*)d_in[1];
  const float* wq    = (const float*)d_in[2];  // (2,128,21,9)
  const float* wk    = (const float*)d_in[3];  // (2,16384,1,9)
  const float* wv    = (const float*)d_in[4];  // (2,16384,1,9)
  const float* wproj = (const float*)d_in[5];  // (2,128,128)
  const float* wh    = (const float*)d_in[6];  // (40,64000)
  const float* bh    = (const float*)d_in[7];  // (40,)
  float* outp = (float*)d_out;
  (void)in_sizes; (void)n_in; (void)out_size; (void)ws_size;

  const size_t NX = (size_t)B_ * C_ * T_;      // 512000 floats
  float* xA = (float*)d_ws;
  float* xB = xA + NX;
  float* Qb = xB + NX;                         // total ws use: 6 MB (16B-aligned)

  hipMemcpyAsync(xA, x, NX * sizeof(float), hipMemcpyDeviceToDevice, stream);

  const float* xcur = xA;
  float* xnxt = xB;
  for (int l = 0; l < 2; ++l) {
    qconv_wmma<<<dim3(32, 8, B_), 32, 0, stream>>>(
        xcur, occ, wq + (size_t)l * D_ * KDIM_, Qb);
    layer_fused<<<dim3(5, C_, B_), 128, 0, stream>>>(
        xcur, Qb,
        wk + (size_t)l * C_ * D_ * KW_,
        wv + (size_t)l * C_ * D_ * KW_,
        wproj + (size_t)l * C_ * D_, xnxt);
    float* tmp = xnxt; xnxt = (float*)xcur; xcur = tmp;
  }
  // after 2 swaps, xcur == xA holds the final x
  head_gemm<<<dim3(40), 256, 0, stream>>>(xcur, wh, bh, outp);
}